// RDC_loss_84164179132681
// MI455X (gfx1250) — compile-verified
//
#include <hip/hip_runtime.h>
#include <math.h>

// ---------------------------------------------------------------------------
// RDC loss for MI455X (gfx1250).
// Memory-bound streaming op: ~545 MB total traffic => ~23us roofline at
// 23.3 TB/s. No matmul exists in this op (gather-dots + matvec reduce), so
// the MI455X-specific optimization is the data path:
//   - Tensor Data Mover (TDM) streams center tiles global->LDS and does the
//     205MB passthrough copy LDS->global without touching VGPRs.
//   - TDM gather-mode descriptor fetches the 8 label-indexed center rows per
//     block in one DMA op.
//   - wave32 butterfly reductions, b128 loads, fixed-order (deterministic)
//     tree reductions, no float atomics.
// Toolchain note: this lane's tensor builtins take the 6-arg form
//   (u32x4 g0, i32x8 g1, i32x4 g2, i32x4 g3, i32x8 pad, i32 cpol).
// ---------------------------------------------------------------------------

typedef float v4f __attribute__((ext_vector_type(4)));
typedef unsigned int u32x4 __attribute__((ext_vector_type(4)));
typedef int i32x4 __attribute__((ext_vector_type(4)));
typedef int i32x8 __attribute__((ext_vector_type(8)));

#define C_CLASSES 100000
#define D_FEAT    512
#define N_BATCH   32768
#define GRID_A    1000
#define ROWS_A    100      // C_CLASSES / GRID_A exactly
#define TILE_R    25       // rows per TDM tile in kernel A (50KB LDS)
#define EPS_N     1e-12f

#if defined(__AMDGCN__) && __has_builtin(__builtin_amdgcn_tensor_load_to_lds) && \
    __has_builtin(__builtin_amdgcn_tensor_store_from_lds) &&                     \
    __has_builtin(__builtin_amdgcn_s_wait_tensorcnt)
#define USE_TDM 1
#else
#define USE_TDM 0
#endif

__device__ __forceinline__ float wave_sum_f32(float v) {
  // wave32 butterfly: all 32 lanes end with the full sum
  #pragma unroll
  for (int off = 16; off > 0; off >>= 1) v += __shfl_xor(v, off, 32);
  return v;
}

#if USE_TDM
// Low 32 bits of a generic pointer to LDS == workgroup-relative LDS offset.
__device__ __forceinline__ unsigned lds_off_u32(const void* p) {
  return (unsigned)(unsigned long long)(uintptr_t)p;
}

// Build a 2D Tensor-DMA descriptor (D# groups 0 and 1).
//   data_size=4B, tensor_dim0 = tile_dim0 = tensor_dim0_stride = 512,
//   tensor_dim1 = tensor_d1, tile_dim1 = tile_d1.
// g0flags: 0 = normal; (1u<<31)|(1u<<30) = gather mode with 32-bit indices.
__device__ __forceinline__ void tdm_desc(unsigned long long gaddr, unsigned ldsoff,
                                         unsigned tensor_d1, unsigned tile_d1,
                                         unsigned g0flags, u32x4* g0, i32x8* g1) {
  u32x4 a;
  a[0] = 1u | g0flags;                                   // count=1 (+gather bits)
  a[1] = ldsoff;                                         // lds_addr
  a[2] = (unsigned)(gaddr & 0xffffffffu);                // global_addr[31:0]
  a[3] = (unsigned)((gaddr >> 32) & 0x1ffffffu)          // global_addr[56:32]
         | (2u << 30);                                   // type=2 ("image")
  i32x8 b;
  b[0] = (int)(2u << 16);                                // data_size = 4 bytes
  b[1] = (int)(512u << 16);                              // tensor_dim0[15:0]=512
  b[2] = (int)((tensor_d1 & 0xffffu) << 16);             // dim0[31:16]=0 | dim1[15:0]
  b[3] = (int)((tensor_d1 >> 16) & 0xffffu)              // tensor_dim1[31:16]
         | (int)(512u << 16);                            // tile_dim0 = 512
  b[4] = (int)(tile_d1 & 0xffffu);                       // tile_dim1 | tile_dim2=0
  b[5] = 512;                                            // tensor_dim0_stride[31:0]
  b[6] = 0;                                              // stride0 hi / stride1 lo
  b[7] = 0;                                              // tensor_dim1_stride hi
  *g0 = a;
  *g1 = b;
}
#endif

// ---------------------------------------------------------------------------
// Kernel A: stream centers once via TDM tiles. Per row (one wave32 per row):
//   sum of squares -> scale = 1/max(||r||,eps); accumulate scale*row into
//   per-wave register column accumulators. Passthrough copy to out_centers is
//   a TDM LDS->global store (dest is only 4B aligned: fine for TDM).
// Per-block partial column sums -> ws (fixed-order cross-wave reduce).
// ---------------------------------------------------------------------------
__global__ __launch_bounds__(256) void rdc_center_pass(
    const float* __restrict__ centers, float* __restrict__ out_centers,
    float* __restrict__ partial) {
#if USE_TDM
  __shared__ float tile[TILE_R * D_FEAT];   // 50KB TDM staging tile
#endif
  __shared__ float acc[8 * D_FEAT];         // one 512-f32 accumulator per wave
  const int wave = threadIdx.x >> 5;
  const int lane = threadIdx.x & 31;
  const int row0 = blockIdx.x * ROWS_A;

  v4f f[4];
  #pragma unroll
  for (int j = 0; j < 4; ++j) f[j] = (v4f){0.f, 0.f, 0.f, 0.f};

#if USE_TDM
  const unsigned tile_lds = lds_off_u32(tile);
  const i32x4 z4 = {0, 0, 0, 0};
  const i32x8 z8 = {0, 0, 0, 0, 0, 0, 0, 0};
  for (int t = 0; t < ROWS_A / TILE_R; ++t) {
    const int trow0 = row0 + t * TILE_R;
    __syncthreads();   // all waves done reading previous tile
    if (wave == 0) {   // single TDM issuer per block (wave-uniform branch)
      // previous passthrough store must finish before LDS is overwritten
      __builtin_amdgcn_s_wait_tensorcnt(0);
      u32x4 g0; i32x8 g1;
      tdm_desc((unsigned long long)(uintptr_t)(centers + (size_t)trow0 * D_FEAT),
               tile_lds, TILE_R, TILE_R, 0u, &g0, &g1);
      __builtin_amdgcn_tensor_load_to_lds(g0, g1, z4, z4, z8, 0);
      __builtin_amdgcn_s_wait_tensorcnt(0);   // tile resident in LDS
      // passthrough copy LDS -> out_centers (TDM store, overlaps compute)
      u32x4 s0; i32x8 s1;
      tdm_desc((unsigned long long)(uintptr_t)(out_centers + (size_t)trow0 * D_FEAT),
               tile_lds, TILE_R, TILE_R, 0u, &s0, &s1);
      __builtin_amdgcn_tensor_store_from_lds(s0, s1, z4, z4, z8, 0);
    }
    __syncthreads();   // tile visible to all waves
    for (int lr = wave; lr < TILE_R; lr += 8) {
      const float* rp = &tile[lr * D_FEAT];
      v4f v[4];
      #pragma unroll
      for (int j = 0; j < 4; ++j)
        v[j] = *(const v4f*)(rp + j * 128 + lane * 4);   // ds_load_b128
      float s = 0.f;
      #pragma unroll
      for (int j = 0; j < 4; ++j)
        #pragma unroll
        for (int tt = 0; tt < 4; ++tt) s = fmaf(v[j][tt], v[j][tt], s);
      s = wave_sum_f32(s);
      const float scl = 1.0f / fmaxf(sqrtf(s), EPS_N);
      #pragma unroll
      for (int j = 0; j < 4; ++j) f[j] += v[j] * scl;
    }
  }
#else
  for (int lr = wave; lr < ROWS_A; lr += 8) {
    const int row = row0 + lr;
    const float* rp = centers + (size_t)row * D_FEAT;
    float* op = out_centers + (size_t)row * D_FEAT;
    v4f v[4];
    #pragma unroll
    for (int j = 0; j < 4; ++j)
      v[j] = *(const v4f*)(rp + j * 128 + lane * 4);
    float s = 0.f;
    #pragma unroll
    for (int j = 0; j < 4; ++j) {
      #pragma unroll
      for (int tt = 0; tt < 4; ++tt) {
        s = fmaf(v[j][tt], v[j][tt], s);
        __builtin_nontemporal_store(v[j][tt], op + j * 128 + lane * 4 + tt);
      }
    }
    s = wave_sum_f32(s);
    const float scl = 1.0f / fmaxf(sqrtf(s), EPS_N);
    #pragma unroll
    for (int j = 0; j < 4; ++j) f[j] += v[j] * scl;
  }
#endif

  #pragma unroll
  for (int j = 0; j < 4; ++j)
    *(v4f*)&acc[wave * D_FEAT + j * 128 + lane * 4] = f[j];
  __syncthreads();

  // fixed-order cross-wave reduce: thread handles dims tid and tid+256
  #pragma unroll
  for (int k = 0; k < 2; ++k) {
    const int d = (int)threadIdx.x + k * 256;
    float s = 0.f;
    #pragma unroll
    for (int w = 0; w < 8; ++w) s += acc[w * D_FEAT + d];
    partial[(size_t)blockIdx.x * D_FEAT + d] = s;
  }
}

// ---------------------------------------------------------------------------
// Kernel B: mean[d] = (sum over GRID_A partials) / C, fixed order.
// ---------------------------------------------------------------------------
__global__ __launch_bounds__(1024) void rdc_mean_reduce(
    const float* __restrict__ partial, float* __restrict__ mean) {
  __shared__ float s2[1024];
  const int d = (int)threadIdx.x & 511;
  const int half = (int)threadIdx.x >> 9;
  const int b0 = half * (GRID_A / 2);
  float s = 0.f;
  for (int b = 0; b < GRID_A / 2; ++b)
    s += partial[(size_t)(b0 + b) * D_FEAT + d];
  s2[threadIdx.x] = s;
  __syncthreads();
  if (threadIdx.x < 512)
    mean[d] = (s2[d] + s2[d + 512]) / (float)C_CLASSES;
}

// ---------------------------------------------------------------------------
// Kernel C: one wave32 per batch row (8 rows/block). The 8 label-indexed
// center rows are fetched with ONE TDM gather-mode descriptor (32-bit row
// indices in D# groups 2/3) into LDS; waves then do four fused dot-reductions
//   sx = x.x, sc = c.c, sxc = x.c, sxm = x.mean
// and the scalar loss term.
// ---------------------------------------------------------------------------
__global__ __launch_bounds__(256) void rdc_batch_pass(
    const float* __restrict__ x, const int* __restrict__ labels,
    const int* __restrict__ epoch_p, const float* __restrict__ centers,
    const float* __restrict__ mean, float* __restrict__ terms) {
  __shared__ float smean[D_FEAT];
#if USE_TDM
  __shared__ float ctile[8 * D_FEAT];   // 16KB: 8 gathered center rows
#endif
  const int wave = threadIdx.x >> 5;
  const int lane = threadIdx.x & 31;
  const int row = blockIdx.x * 8 + wave;

  for (int i = threadIdx.x; i < D_FEAT; i += 256) smean[i] = mean[i];

#if USE_TDM
  if (wave == 0) {   // single TDM issuer (wave-uniform branch)
    i32x4 idxA, idxB;   // 8 gather row indices = this block's labels (uniform)
    #pragma unroll
    for (int w = 0; w < 4; ++w) idxA[w] = labels[blockIdx.x * 8 + w];
    #pragma unroll
    for (int w = 0; w < 4; ++w) idxB[w] = labels[blockIdx.x * 8 + 4 + w];
    const i32x8 z8 = {0, 0, 0, 0, 0, 0, 0, 0};
    u32x4 g0; i32x8 g1;
    tdm_desc((unsigned long long)(uintptr_t)centers, lds_off_u32(ctile),
             C_CLASSES, /*tile_dim1 = #valid indices*/ 8u,
             /*gather_mode | 32-bit indices*/ (1u << 31) | (1u << 30), &g0, &g1);
    __builtin_amdgcn_tensor_load_to_lds(g0, g1, idxA, idxB, z8, 0);
    __builtin_amdgcn_s_wait_tensorcnt(0);
  }
  __syncthreads();
  const float* cp = &ctile[wave * D_FEAT];
#else
  __syncthreads();
  const int lab = labels[row];
  const float* cp = centers + (size_t)lab * D_FEAT;
  if (lane < 16) __builtin_prefetch((const void*)(cp + lane * 32), 0, 0);
#endif

  const float* xp = x + (size_t)row * D_FEAT;

  v4f xv[4], cv[4], mv[4];
  #pragma unroll
  for (int j = 0; j < 4; ++j) {
    const int e = j * 128 + lane * 4;
    cv[j] = *(const v4f*)(cp + e);       // LDS (TDM path) or global gather
    xv[j] = *(const v4f*)(xp + e);       // streaming b128
    mv[j] = *(const v4f*)&smean[e];      // LDS b128
  }

  float sx = 0.f, sc = 0.f, sxc = 0.f, sxm = 0.f;
  #pragma unroll
  for (int j = 0; j < 4; ++j) {
    #pragma unroll
    for (int t = 0; t < 4; ++t) {
      const float xf = xv[j][t], cf = cv[j][t], mf = mv[j][t];
      sx  = fmaf(xf, xf, sx);
      sc  = fmaf(cf, cf, sc);
      sxc = fmaf(xf, cf, sxc);
      sxm = fmaf(xf, mf, sxm);
    }
  }
  sx  = wave_sum_f32(sx);
  sc  = wave_sum_f32(sc);
  sxc = wave_sum_f32(sxc);
  sxm = wave_sum_f32(sxm);

  if (lane == 0) {
    const float nx = fmaxf(sqrtf(sx), EPS_N);
    const float nc = fmaxf(sqrtf(sc), EPS_N);
    const float s_all = sxm / nx;
    const float s_lab = sxc / (nx * nc);
    const float e_all = expf(s_all);
    const float kC = (float)(((double)C_CLASSES + 1.0) / (double)C_CLASSES);
    const float t1 = (e_all - expf(s_lab * kC)) / 1.0f;            // ALPHA1
    const float t2 = -fabsf((e_all - expf(s_lab) + 0.2f) / 50.0f); // ALPHA2
    const float vmix = fminf((float)(*epoch_p) / 14.0f, 1.0f);     // WARMUP-1
    terms[row] = (1.0f - vmix) * t1 + vmix * t2;
  }
}

// ---------------------------------------------------------------------------
// Kernel D: deterministic 32768 -> 1 reduction, /B, write loss to d_out[0].
// ---------------------------------------------------------------------------
__global__ __launch_bounds__(1024) void rdc_final_reduce(
    const float* __restrict__ terms, float* __restrict__ out) {
  __shared__ float s[1024];
  float a = 0.f;
  #pragma unroll
  for (int i = 0; i < N_BATCH / 1024; ++i)
    a += terms[threadIdx.x + i * 1024];
  s[threadIdx.x] = a;
  __syncthreads();
  for (int off = 512; off > 0; off >>= 1) {
    if ((int)threadIdx.x < off) s[threadIdx.x] += s[threadIdx.x + off];
    __syncthreads();
  }
  if (threadIdx.x == 0) out[0] = s[0] / (float)N_BATCH;
}

extern "C" void kernel_launch(void* const* d_in, const int* in_sizes, int n_in,
                              void* d_out, int out_size, void* d_ws, size_t ws_size,
                              hipStream_t stream) {
  (void)in_sizes; (void)n_in; (void)out_size; (void)ws_size;
  const float* x       = (const float*)d_in[0];
  const int*   labels  = (const int*)d_in[1];
  /* d_in[2] = ori_labels: unused by the reference computation */
  const int*   epoch   = (const int*)d_in[3];
  const float* centers = (const float*)d_in[4];

  float* out         = (float*)d_out;       // [0] = loss, [1..] = centers copy
  float* out_centers = out + 1;

  float* ws      = (float*)d_ws;
  float* partial = ws;                              // GRID_A * 512 floats
  float* mean    = ws + (size_t)GRID_A * D_FEAT;    // 512 floats
  float* terms   = mean + D_FEAT;                   // N_BATCH floats

  rdc_center_pass <<<GRID_A,        256, 0, stream>>>(centers, out_centers, partial);
  rdc_mean_reduce <<<1,            1024, 0, stream>>>(partial, mean);
  rdc_batch_pass  <<<N_BATCH / 8,   256, 0, stream>>>(x, labels, epoch, centers, mean, terms);
  rdc_final_reduce<<<1,            1024, 0, stream>>>(terms, out);
}